// SumVector_21818433863919
// MI455X (gfx1250) — compile-verified
//
#include <hip/hip_runtime.h>
#include <math.h>

typedef __attribute__((ext_vector_type(2))) float v2f;
typedef __attribute__((ext_vector_type(8))) float v8f;

#define BATCH 2048
#define NIN   1024
#define NSUM  256

#define BM 128
#define BN 32
#define KC 32
#define LDA 36   // padded LDS stride for A tile: conflict-free fragment reads
#define LDB 40   // padded LDS stride for B tile: 2*LDB % 64 == 16 separates K-halves

// ---------------------------------------------------------------------------
// Kernel A: per-column stats of params [NIN x NSUM] row-major.
// Mj[j] = max_i params[i][j]; Cj[j] = Mj - logsumexp_i = -log(sum exp(v-Mj)).
// One block of NSUM threads; thread j walks its column (coalesced across j).
// ---------------------------------------------------------------------------
__global__ void col_stats_kernel(const float* __restrict__ params,
                                 float* __restrict__ Mj,
                                 float* __restrict__ Cj) {
  const int j = threadIdx.x;
  float m = -INFINITY;
  float s = 0.0f;
  for (int i = 0; i < NIN; ++i) {
    float v = params[i * NSUM + j];
    if (v > m) {
      s = s * __expf(m - v) + 1.0f;
      m = v;
    } else {
      s += __expf(v - m);
    }
  }
  Mj[j] = m;
  Cj[j] = -__logf(s);
}

// ---------------------------------------------------------------------------
// Kernel B: per-row max of prods [BATCH x NIN]; one block per row.
// ---------------------------------------------------------------------------
__global__ void row_max_kernel(const float* __restrict__ prods,
                               float* __restrict__ mb) {
  __shared__ float red[256];
  const int row = blockIdx.x;
  const float* p = prods + (size_t)row * NIN;
  float m = -INFINITY;
  for (int i = threadIdx.x; i < NIN; i += 256) m = fmaxf(m, p[i]);
  red[threadIdx.x] = m;
  __syncthreads();
  for (int off = 128; off > 0; off >>= 1) {
    if (threadIdx.x < off)
      red[threadIdx.x] = fmaxf(red[threadIdx.x], red[threadIdx.x + off]);
    __syncthreads();
  }
  if (threadIdx.x == 0) mb[row] = red[0];
}

// ---------------------------------------------------------------------------
// Kernel C: (+,logsumexp)-semiring matmul as fp32 WMMA GEMM on exp-shifted
// operands, with exp applied on the fly while staging tiles into LDS.
//   S[b][j] = sum_i exp(prods[b][i]-mb[b]) * exp(params[i][j]-Mj[j])
//   out[b][j] = log(S) + mb[b] + Cj[j]
// Block: 256 threads = 8 waves as 4(M) x 2(N); wave tile 32x16 (2 WMMA accs).
// ---------------------------------------------------------------------------
__global__ __launch_bounds__(256)
void lse_gemm_kernel(const float* __restrict__ prods,
                     const float* __restrict__ params,
                     const float* __restrict__ mb,
                     const float* __restrict__ Mj,
                     const float* __restrict__ Cj,
                     float* __restrict__ out) {
  __shared__ float lsA[BM * LDA];  // 128 x 32 (padded)
  __shared__ float lsB[KC * LDB];  // 32 x 32  (padded)

  const int tid   = threadIdx.x;
  const int lane  = tid & 31;
  const int wv    = tid >> 5;       // 0..7
  const int waveM = wv & 3;         // M offset = 32*waveM
  const int waveN = wv >> 2;        // N offset = 16*waveN
  const int m0 = blockIdx.x * BM;
  const int n0 = blockIdx.y * BN;

  // --- staging thread coordinates: each thread moves one float4 per row-pass
  const int ldRow = tid >> 3;        // 0..31
  const int ldCol = (tid & 7) * 4;   // 0,4,...,28

  // Hoisted per-thread constants for the on-the-fly exp transforms
  float rowShift[4];
#pragma unroll
  for (int pass = 0; pass < 4; ++pass)
    rowShift[pass] = mb[m0 + ldRow + pass * 32];
  const float mj0 = Mj[n0 + ldCol + 0];
  const float mj1 = Mj[n0 + ldCol + 1];
  const float mj2 = Mj[n0 + ldCol + 2];
  const float mj3 = Mj[n0 + ldCol + 3];

  v8f acc0 = {};
  v8f acc1 = {};

  const int fm = lane & 15;          // fragment row (A) / col (B)
  const int kh = (lane >> 4) << 1;   // K-half offset: 0 or 2

  for (int k0 = 0; k0 < NIN; k0 += KC) {
    // ---- stage A tile: exp(prods - mb), 4 passes of 32 rows, float4 loads
#pragma unroll
    for (int pass = 0; pass < 4; ++pass) {
      const int r = ldRow + pass * 32;
      const float4 va = *(const float4*)(prods + (size_t)(m0 + r) * NIN + k0 + ldCol);
      const float rm = rowShift[pass];
      float* dst = &lsA[r * LDA + ldCol];
      dst[0] = __expf(va.x - rm);
      dst[1] = __expf(va.y - rm);
      dst[2] = __expf(va.z - rm);
      dst[3] = __expf(va.w - rm);
    }
    // ---- stage B tile: exp(params - Mj), 32 K-rows in one pass
    {
      const float4 vb = *(const float4*)(params + (size_t)(k0 + ldRow) * NSUM + n0 + ldCol);
      float* dst = &lsB[ldRow * LDB + ldCol];
      dst[0] = __expf(vb.x - mj0);
      dst[1] = __expf(vb.y - mj1);
      dst[2] = __expf(vb.z - mj2);
      dst[3] = __expf(vb.w - mj3);
    }
    __syncthreads();

    // ---- WMMA over the staged K chunk (K=4 per op)
#pragma unroll
    for (int kk = 0; kk < KC; kk += 4) {
      v2f a0, a1, b;
      const float* pa0 = &lsA[(waveM * 32 + fm) * LDA + kk + kh];
      a0.x = pa0[0];
      a0.y = pa0[1];
      const float* pa1 = &lsA[(waveM * 32 + 16 + fm) * LDA + kk + kh];
      a1.x = pa1[0];
      a1.y = pa1[1];
      const float* pb = &lsB[(kk + kh) * LDB + waveN * 16 + fm];
      b.x = pb[0];
      b.y = pb[LDB];
      acc0 = __builtin_amdgcn_wmma_f32_16x16x4_f32(
          false, a0, false, b, (short)0, acc0, false, false);
      acc1 = __builtin_amdgcn_wmma_f32_16x16x4_f32(
          false, a1, false, b, (short)0, acc1, false, false);
    }
    __syncthreads();
  }

  // ---- epilogue: out = log(acc) + mb[row] + Cj[col]
  const int n   = lane & 15;
  const int khv = lane >> 4;  // 0/1 selects M = r or r+8
  const int col = n0 + waveN * 16 + n;
  const float cj = Cj[col];
#pragma unroll
  for (int r = 0; r < 8; ++r) {
    const int row0 = m0 + waveM * 32 + r + 8 * khv;
    out[(size_t)row0 * NSUM + col] = __logf(acc0[r]) + mb[row0] + cj;
    const int row1 = row0 + 16;
    out[(size_t)row1 * NSUM + col] = __logf(acc1[r]) + mb[row1] + cj;
  }
}

// ---------------------------------------------------------------------------
extern "C" void kernel_launch(void* const* d_in, const int* in_sizes, int n_in,
                              void* d_out, int out_size, void* d_ws, size_t ws_size,
                              hipStream_t stream) {
  const float* prods  = (const float*)d_in[0];   // [BATCH, NIN]
  const float* params = (const float*)d_in[1];   // [NIN, NSUM]
  float* mb = (float*)d_ws;          // BATCH floats
  float* Mj = mb + BATCH;            // NSUM floats
  float* Cj = Mj + NSUM;             // NSUM floats
  float* out = (float*)d_out;        // [BATCH, NSUM]

  col_stats_kernel<<<1, NSUM, 0, stream>>>(params, Mj, Cj);
  row_max_kernel<<<BATCH, 256, 0, stream>>>(prods, mb);

  dim3 grid(BATCH / BM, NSUM / BN);  // 16 x 8 = 128 workgroups
  lse_gemm_kernel<<<grid, 256, 0, stream>>>(prods, params, mb, Mj, Cj, out);
}